// MFA_40690520162819
// MI455X (gfx1250) — compile-verified
//
#include <hip/hip_runtime.h>
#include <math.h>

typedef __attribute__((ext_vector_type(2))) float v2f;
typedef __attribute__((ext_vector_type(8))) float v8f;
typedef unsigned int u32x4 __attribute__((ext_vector_type(4)));
typedef int i32x8 __attribute__((ext_vector_type(8)));
typedef int i32x4 __attribute__((ext_vector_type(4)));

#define KK 32
#define DD 256
#define QQ 16
#define NN 4096
#define LOG2PI_F 1.8378770664093453f

// per-component workspace layout (in floats)
#define PKF       4880
#define OFF_MUISP 0      // float2[256]: {mu, 1/sqrt(diag)}
#define OFF_WT    512    // Wt[16][256]  (W^T, row-major)
#define OFF_MINV  4608   // Minv[16][16]
#define OFF_SCAL  4864   // [0]=logdet C, [1]=log_pi

// ---------------------------------------------------------------------------
// Phase 1: per-component Woodbury precompute (tiny; grid=K, block=256)
// ---------------------------------------------------------------------------
__global__ void __launch_bounds__(256)
mfa_prep(const float* __restrict__ log_pi,
         const float* __restrict__ mu,
         const float* __restrict__ Lambda,
         const float* __restrict__ log_psi,
         float* __restrict__ ws) {
  const int k = blockIdx.x;
  const int d = threadIdx.x;           // 0..255 == D index
  float* wk = ws + (size_t)k * PKF;

  __shared__ float Wt_s[QQ * DD];      // 16 KB
  __shared__ float red[DD];
  __shared__ float M_s[QQ * QQ];
  __shared__ float Rinv_s[QQ * QQ];

  const float diag = __expf(log_psi[k * DD + d]) + 1e-6f + 1e-5f;
  const float isp  = rsqrtf(diag);
  wk[OFF_MUISP + 2 * d]     = mu[k * DD + d];
  wk[OFF_MUISP + 2 * d + 1] = isp;
  red[d] = __logf(diag);

#pragma unroll
  for (int q = 0; q < QQ; ++q) {
    const float w = Lambda[((size_t)k * DD + d) * QQ + q] * isp;
    Wt_s[q * DD + d]        = w;
    wk[OFF_WT + q * DD + d] = w;
  }
  __syncthreads();

  // sum_d log(diag)
  for (int s = DD / 2; s > 0; s >>= 1) {
    if (d < s) red[d] += red[d + s];
    __syncthreads();
  }

  // M = I + W^T W   (16x16, one element per thread)
  {
    const int i = d >> 4, j = d & 15;
    float acc = (i == j) ? 1.0f : 0.0f;
    for (int p = 0; p < DD; ++p) acc += Wt_s[i * DD + p] * Wt_s[j * DD + p];
    M_s[i * QQ + j] = acc;
  }
  __syncthreads();

  if (d == 0) {
    // in-place lower Cholesky of M
    float ld2 = 0.0f;
    for (int i = 0; i < QQ; ++i) {
      for (int j = 0; j <= i; ++j) {
        float s = M_s[i * QQ + j];
        for (int p = 0; p < j; ++p) s -= M_s[i * QQ + p] * M_s[j * QQ + p];
        if (j == i) {
          const float l = sqrtf(s);
          M_s[i * QQ + i] = l;
          ld2 += __logf(l);
        } else {
          M_s[i * QQ + j] = s / M_s[j * QQ + j];
        }
      }
    }
    // invert lower-triangular L
    for (int i = 0; i < QQ; ++i) {
      Rinv_s[i * QQ + i] = 1.0f / M_s[i * QQ + i];
      for (int j = 0; j < i; ++j) {
        float s = 0.0f;
        for (int p = j; p < i; ++p) s += M_s[i * QQ + p] * Rinv_s[p * QQ + j];
        Rinv_s[i * QQ + j] = -s / M_s[i * QQ + i];
      }
      for (int j = i + 1; j < QQ; ++j) Rinv_s[i * QQ + j] = 0.0f;
    }
    // Minv = Rinv^T Rinv
    for (int i = 0; i < QQ; ++i)
      for (int j = 0; j < QQ; ++j) {
        float s = 0.0f;
        for (int q = 0; q < QQ; ++q) s += Rinv_s[q * QQ + i] * Rinv_s[q * QQ + j];
        wk[OFF_MINV + i * QQ + j] = s;
      }
    wk[OFF_SCAL]     = red[0] + 2.0f * ld2;  // log det C_k
    wk[OFF_SCAL + 1] = log_pi[k];
  }
}

// ---------------------------------------------------------------------------
// Phase 2: main pass. grid=(N/64, K), block=128 (4 waves).
// W^T tile (16KB) staged into LDS with the Tensor Data Mover; GEMM via
// v_wmma_f32_16x16x4_f32; ||u||^2 in packed VALU alongside; M^-1 quadratic
// form via LDS exchange + shuffles.
// ---------------------------------------------------------------------------
__global__ void __launch_bounds__(128)
mfa_estep(const float* __restrict__ X,
          const float* __restrict__ ws,
          float* __restrict__ out) {
  const int k   = blockIdx.y;
  const float* wk = ws + (size_t)k * PKF;
  const int tid  = threadIdx.x;
  const int wave = tid >> 5;
  const int lane = tid & 31;
  const int col  = lane & 15;   // N column within tile / q row for A
  const int half = lane >> 4;   // 0: K-offsets {0,1}, 1: {2,3}

  __shared__ float2 muisp_s[DD];       // 2 KB
  __shared__ float  Wt_s[QQ * DD];     // 16 KB  (filled by TDM)
  __shared__ float  Minv_s[QQ * QQ];   // 1 KB
  __shared__ float  texch[4][QQ * QQ]; // 4 KB, per-wave t exchange
  __shared__ float  s_ld, s_lpi;

  // --- TDM: DMA the 16x256 f32 W^T tile from ws into LDS (wave 0 issues) ---
  if (wave == 0) {
    u32x4 g0;
    g0.x = 1u;                                        // count=1, user mode
    g0.y = (unsigned int)(size_t)(void*)Wt_s;         // lds_addr (byte offset)
    unsigned long long ga = (unsigned long long)(size_t)(const void*)(wk + OFF_WT);
    g0.z = (unsigned int)ga;                          // global_addr[31:0]
    g0.w = (unsigned int)((ga >> 32) & 0x01FFFFFFull) // global_addr[56:32]
           | 0x80000000u;                             // type=2 ("image")
    i32x8 g1;
    g1[0] = 0x00020000;   // workgroup_mask=0, data_size=2 (4 bytes)
    g1[1] = 0x01000000;   // tensor_dim0[15:0]=256 in bits[63:48]
    g1[2] = 0x00100000;   // tensor_dim1[15:0]=16  in bits[95:80]
    g1[3] = 0x01000000;   // tile_dim0=256         in bits[127:112]
    g1[4] = 0x00000010;   // tile_dim1=16, tile_dim2=0
    g1[5] = 256;          // tensor_dim0_stride[31:0]=256
    g1[6] = 0;
    g1[7] = 0;
    i32x4 gz4 = {0, 0, 0, 0};              // groups 2/3: tensor is 2D
    i32x8 gz8 = {0, 0, 0, 0, 0, 0, 0, 0};
    __builtin_amdgcn_tensor_load_to_lds(g0, g1, gz4, gz4, gz8, 0);
  }

  // small tables staged by all threads meanwhile
  const float2* muisp_g = (const float2*)(wk + OFF_MUISP);
  for (int i = tid; i < DD; i += 128)      muisp_s[i] = muisp_g[i];
  for (int i = tid; i < QQ * QQ; i += 128) Minv_s[i]  = wk[OFF_MINV + i];
  if (tid == 0) { s_ld = wk[OFF_SCAL]; s_lpi = wk[OFF_SCAL + 1]; }

  if (wave == 0) __builtin_amdgcn_s_wait_tensorcnt(0);
  __syncthreads();

  const int n = (blockIdx.x * 4 + wave) * 16 + col;
  const float* xr = X + (size_t)n * DD;
  __builtin_prefetch(xr, 0, 1);   // global_prefetch_b8

  v8f acc = {};
  float q2 = 0.0f;

#pragma unroll 4
  for (int d0 = 0; d0 < DD; d0 += 4) {
    const int d = d0 + 2 * half;
    // A (16x4 chunk of W^T): lane row = q = col, K split across halves
    v2f a;
    a.x = Wt_s[col * DD + d];
    a.y = Wt_s[col * DD + d + 1];
    // B (4x16 chunk of whitened diff u): lane column = n = col
    const float2 x2 = *(const float2*)(xr + d);
    const float2 m0 = muisp_s[d];
    const float2 m1 = muisp_s[d + 1];
    v2f b;
    b.x = (x2.x - m0.x) * m0.y;
    b.y = (x2.y - m1.x) * m1.y;
    q2 += b.x * b.x + b.y * b.y;
    acc = __builtin_amdgcn_wmma_f32_16x16x4_f32(false, a, false, b,
                                                (short)0, acc, false, false);
  }
  // fold the two half-lane partial ||u||^2 contributions per column
  q2 += __shfl_xor(q2, 16, 32);

  // acc[v] = t[q][n], q = v + 8*half, n = col. Exchange t through LDS.
  float* tb = texch[wave];
#pragma unroll
  for (int v = 0; v < 8; ++v) tb[(v + 8 * half) * QQ + col] = acc[v];
  __syncthreads();

  float tcol[QQ];
#pragma unroll
  for (int q = 0; q < QQ; ++q) tcol[q] = tb[q * QQ + col];

  // correction = t^T Minv t ; each half-lane covers its 8 rows of t
  float corr = 0.0f;
#pragma unroll
  for (int v = 0; v < 8; ++v) {
    const int q = v + 8 * half;
    float s = 0.0f;
#pragma unroll
    for (int j = 0; j < QQ; ++j) s += Minv_s[q * QQ + j] * tcol[j];
    corr += acc[v] * s;
  }
  corr += __shfl_xor(corr, 16, 32);

  if (half == 0) {
    const float maha = q2 - corr;
    const float lp = -0.5f * (maha + (float)DD * LOG2PI_F + s_ld);
    out[(size_t)n * KK + k] = lp + s_lpi;  // unnormalized log-resp
  }
}

// ---------------------------------------------------------------------------
// Phase 3: logsumexp over K=32 (one wave32 per row), normalize in place,
// write log_likelihood to tail of d_out.
// ---------------------------------------------------------------------------
__global__ void __launch_bounds__(256)
mfa_finalize(float* __restrict__ out) {
  const int tid  = threadIdx.x;
  const int wave = tid >> 5;
  const int lane = tid & 31;              // lane == component k
  const int n = blockIdx.x * 8 + wave;

  const float v = out[(size_t)n * KK + lane];
  float m = v;
#pragma unroll
  for (int off = 16; off > 0; off >>= 1) m = fmaxf(m, __shfl_xor(m, off, 32));
  float s = __expf(v - m);
#pragma unroll
  for (int off = 16; off > 0; off >>= 1) s += __shfl_xor(s, off, 32);
  const float lse = m + __logf(s);

  out[(size_t)n * KK + lane] = v - lse;
  if (lane == 0) out[(size_t)NN * KK + n] = lse;
}

// ---------------------------------------------------------------------------
extern "C" void kernel_launch(void* const* d_in, const int* in_sizes, int n_in,
                              void* d_out, int out_size, void* d_ws, size_t ws_size,
                              hipStream_t stream) {
  const float* X       = (const float*)d_in[0];
  const float* log_pi  = (const float*)d_in[1];
  const float* mu      = (const float*)d_in[2];
  const float* Lambda  = (const float*)d_in[3];
  const float* log_psi = (const float*)d_in[4];
  float* out = (float*)d_out;
  float* ws  = (float*)d_ws;

  mfa_prep<<<KK, DD, 0, stream>>>(log_pi, mu, Lambda, log_psi, ws);
  mfa_estep<<<dim3(NN / 64, KK), 128, 0, stream>>>(X, ws, out);
  mfa_finalize<<<NN / 8, 256, 0, stream>>>(out);
}